// BlockVQ_18468359373179
// MI455X (gfx1250) — compile-verified
//
#include <hip/hip_runtime.h>
#include <hip/hip_bf16.h>

#if defined(__HIP_DEVICE_COMPILE__)
#if !__has_builtin(__builtin_amdgcn_wmma_f32_16x16x4_f32)
#error "gfx1250 f32 WMMA builtin not available on device"
#endif
#endif

typedef __attribute__((ext_vector_type(2))) float v2f;
typedef __attribute__((ext_vector_type(8))) float v8f;
typedef int b128_t __attribute__((vector_size(16)));   // matches builtin param type

namespace {
constexpr int kBL   = 8;
constexpr int kDim  = 64;                    // BL*BL
constexpr int kK    = 1024;                  // codes per channel
constexpr int kC    = 16;
constexpr int kB    = 16;
constexpr int kH    = 256;
constexpr int kW    = 256;
constexpr int kNblk = (kH / kBL) * (kW / kBL);     // 1024 blocks per (b,c)
constexpr int kM    = kB * kNblk;                  // 16384 rows per channel
constexpr int kWavesPerWG  = 8;
constexpr int kRowsPerWave = 16;
constexpr int kRowsPerWG   = kWavesPerWG * kRowsPerWave;  // 128
constexpr int kKTile  = 16;
constexpr int kKTiles = kK / kKTile;               // 64
}

// ---- CDNA5 async global->LDS copy (bypasses VGPRs, tracked by ASYNCcnt) ----
__device__ __forceinline__ void tile_fill_16B(const float* gsrc, float* ldst)
{
#if defined(__HIP_DEVICE_COMPILE__) && \
    __has_builtin(__builtin_amdgcn_global_load_async_to_lds_b128)
    __builtin_amdgcn_global_load_async_to_lds_b128(
        (__attribute__((address_space(1))) b128_t*)gsrc,
        (__attribute__((address_space(3))) b128_t*)ldst,
        /*offset=*/0, /*cpol=*/0);
#else
    *(float4*)ldst = *(const float4*)gsrc;   // fallback: VGPR staging
#endif
}

__device__ __forceinline__ void tile_fill_wait()
{
#if defined(__HIP_DEVICE_COMPILE__) && \
    __has_builtin(__builtin_amdgcn_global_load_async_to_lds_b128) && \
    __has_builtin(__builtin_amdgcn_s_wait_asynccnt)
    __builtin_amdgcn_s_wait_asynccnt(0);
#endif
}

// ---------------- prep: codebook norms + zero accumulators ----------------
__global__ __launch_bounds__(256)
void vq_prep(const float* __restrict__ cb,
             float* __restrict__ cbnorm,
             unsigned int* __restrict__ counts,
             float* __restrict__ lossAcc)
{
    const int t = blockIdx.x * blockDim.x + threadIdx.x;
    if (t < kC * kK) {
        const float* p = cb + (size_t)t * kDim;
        float s = 0.f;
        #pragma unroll
        for (int i = 0; i < kDim; i += 2) {
            const v2f v = *(const v2f*)(p + i);
            s += v.x * v.x + v.y * v.y;
        }
        cbnorm[t] = s;
        counts[t] = 0u;
        if (t < kC) lossAcc[t] = 0.f;
    }
}

// ---------------- main: distances via f32 WMMA + argmin + writeback -------
__global__ __launch_bounds__(256)
void vq_main(const float* __restrict__ x,
             const float* __restrict__ cb,
             const float* __restrict__ cbnorm,
             unsigned int* __restrict__ counts,
             float* __restrict__ lossAcc,
             float* __restrict__ xhat,
             float* __restrict__ idxOut)
{
    __shared__ float sCB[2][kKTile * kDim];          // double-buffered codebook tile
    __shared__ float sVN[kWavesPerWG][kRowsPerWave]; // per-row ||v||^2
    __shared__ int   sIDX[kWavesPerWG][kRowsPerWave];

    const int c    = blockIdx.y;
    const int tid  = threadIdx.x;
    const int wave = tid >> 5;
    const int lane = tid & 31;
    const int half = lane >> 4;
    const int lp   = lane & 15;
    const int m0   = blockIdx.x * kRowsPerWG + wave * kRowsPerWave;

    // preload first codebook tile via async copy (4 KB, 256 threads x b128)
    const float* cbc = cb + (size_t)c * kK * kDim;
    tile_fill_16B(cbc + tid * 4, &sCB[0][tid * 4]);

    // ---- load A tile (16 rows x 64 dims) in WMMA 16x16x4 A layout ----
    // lanes 0-15: row=lp, K pair {4s, 4s+1}; lanes 16-31: row=lp, K pair {4s+2, 4s+3}
    const int rowA = m0 + lp;
    const int biA  = rowA >> 10;
    const int nbA  = rowA & 1023;
    const float* vbase = x + (((size_t)biA * kC + c) * kH + (nbA >> 5) * kBL) * kW
                           + (nbA & 31) * kBL;

    v2f a[16];
    float pn = 0.f;
    #pragma unroll
    for (int s = 0; s < 16; ++s) {
        const int d0 = 4 * s + 2 * half;
        const v2f av = *(const v2f*)(vbase + (d0 >> 3) * kW + (d0 & 7));
        a[s] = av;
        pn += av.x * av.x + av.y * av.y;
    }
    const float vnorm = pn + __shfl_xor(pn, 16, 32);
    if (half == 0) sVN[wave][lp] = vnorm;

    tile_fill_wait();          // my async fill of sCB[0] has landed
    __syncthreads();           // everyone's fill + sVN visible

    float vn[8];
    #pragma unroll
    for (int i = 0; i < 8; ++i) vn[i] = sVN[wave][i + 8 * half];

    float best[8];
    int   bidx[8];
    #pragma unroll
    for (int i = 0; i < 8; ++i) { best[i] = 3.4e38f; bidx[i] = 0; }

    int buf = 0;
    for (int kt = 0; kt < kKTiles; ++kt) {
        if (kt + 1 < kKTiles) {  // async-prefetch next tile into the other buffer
            tile_fill_16B(cbc + (size_t)(kt + 1) * kKTile * kDim + tid * 4,
                          &sCB[buf ^ 1][tid * 4]);
        }
        const float* tile = sCB[buf];
        const float  cbn  = cbnorm[c * kK + kt * kKTile + lp];

        v8f acc = {};
        #pragma unroll
        for (int s = 0; s < 16; ++s) {
            // B 4x16 layout mirrored from A: lane lp = column, half selects K pair
            const v2f bv = *(const v2f*)(tile + lp * kDim + 4 * s + 2 * half);
            acc = __builtin_amdgcn_wmma_f32_16x16x4_f32(false, a[s], false, bv,
                                                        (short)0, acc, false, false);
        }
        // acc[i] = dot(row m0 + i + 8*half, code kt*16 + lp)
        #pragma unroll
        for (int i = 0; i < 8; ++i) {
            const float d2  = vn[i] - 2.0f * acc[i] + cbn;
            const int   col = kt * kKTile + lp;
            if (d2 < best[i]) { best[i] = d2; bidx[i] = col; }
        }
        tile_fill_wait();      // next-buffer async fill complete (this wave)
        __syncthreads();       // all waves done reading buf / filling buf^1
        buf ^= 1;
    }

    // argmin reduce across the 16 lanes of each half (tie-break: lowest index)
    #pragma unroll
    for (int off = 1; off < 16; off <<= 1) {
        #pragma unroll
        for (int i = 0; i < 8; ++i) {
            const float ov = __shfl_xor(best[i], off, 32);
            const int   oi = __shfl_xor(bidx[i], off, 32);
            if (ov < best[i] || (ov == best[i] && oi < bidx[i])) {
                best[i] = ov; bidx[i] = oi;
            }
        }
    }
    if (lp == 0) {
        #pragma unroll
        for (int i = 0; i < 8; ++i) {
            const int r   = i + 8 * half;
            const int row = m0 + r;
            const int bi  = row >> 10;
            const int nb  = row & 1023;
            sIDX[wave][r] = bidx[i];
            idxOut[((size_t)bi * kC + c) * kNblk + nb] = (float)bidx[i];
            atomicAdd(&counts[c * kK + bidx[i]], 1u);
        }
    }
    __syncthreads();

    // ---- write x_hat (= chosen code vector) + per-channel MSE loss ----
    float lsum = 0.f;
    #pragma unroll 4
    for (int r = 0; r < kRowsPerWave; ++r) {
        const int row = m0 + r;
        const int bi  = row >> 10;
        const int nb  = row & 1023;
        const size_t base = (((size_t)bi * kC + c) * kH + (nb >> 5) * kBL) * kW
                            + (nb & 31) * kBL;
        const float* q  = cbc + (size_t)sIDX[wave][r] * kDim;
        const int dd  = 2 * lane;                       // each lane: 2 of 64 dims
        const int off = (dd >> 3) * kW + (dd & 7);
        const v2f qv = *(const v2f*)(q + dd);
        const v2f vv = *(const v2f*)(x + base + off);
        *(v2f*)(xhat + base + off) = qv;
        const float e0 = qv.x - vv.x, e1 = qv.y - vv.y;
        lsum += e0 * e0 + e1 * e1;
    }
    #pragma unroll
    for (int off = 16; off > 0; off >>= 1) lsum += __shfl_xor(lsum, off, 32);
    if (lane == 0) atomicAdd(&lossAcc[c], lsum);
}

// ---------------- finalize: loss scaling + perplexity ----------------
__global__ __launch_bounds__(256)
void vq_finalize(const unsigned int* __restrict__ counts,
                 const float* __restrict__ lossAcc,
                 float* __restrict__ lossOut,
                 float* __restrict__ perpOut)
{
    __shared__ float red[8];
    const int c = blockIdx.x;
    const int t = threadIdx.x;
    float s = 0.f;
    for (int k = t; k < kK; k += 256) {
        const float p = (float)counts[c * kK + k] * (1.0f / 16384.0f);
        s += p * __logf(p + 1e-10f);
    }
    #pragma unroll
    for (int off = 16; off > 0; off >>= 1) s += __shfl_xor(s, off, 32);
    if ((t & 31) == 0) red[t >> 5] = s;
    __syncthreads();
    if (t == 0) {
        float tot = 0.f;
        #pragma unroll
        for (int i = 0; i < 8; ++i) tot += red[i];
        perpOut[c] = __expf(-tot);
        lossOut[c] = lossAcc[c] * (1.0f / (16384.0f * 64.0f));
    }
}

extern "C" void kernel_launch(void* const* d_in, const int* in_sizes, int n_in,
                              void* d_out, int out_size, void* d_ws, size_t ws_size,
                              hipStream_t stream)
{
    const float* x  = (const float*)d_in[0];   // (16,16,256,256) f32
    const float* cb = (const float*)d_in[1];   // (16,1024,64) f32

    float* out     = (float*)d_out;
    float* xhat    = out;                                       // 16,777,216
    float* idxOut  = out + (size_t)kB * kC * kH * kW;           // +262,144
    float* lossOut = idxOut + (size_t)kB * kC * kNblk;          // +16
    float* perpOut = lossOut + kC;                              // +16

    char* ws = (char*)d_ws;
    float*        cbnorm  = (float*)ws;                                  // 64 KB
    unsigned int* counts  = (unsigned int*)(ws + (size_t)kC * kK * 4);   // 64 KB
    float*        lossAcc = (float*)(ws + (size_t)2 * kC * kK * 4);      // 64 B

    vq_prep<<<dim3((kC * kK + 255) / 256), dim3(256), 0, stream>>>(
        cb, cbnorm, counts, lossAcc);

    vq_main<<<dim3(kM / kRowsPerWG, kC), dim3(256), 0, stream>>>(
        x, cb, cbnorm, counts, lossAcc, xhat, idxOut);

    vq_finalize<<<dim3(kC), dim3(256), 0, stream>>>(
        counts, lossAcc, lossOut, perpOut);
}